// SimilarityLoss_71923522339276
// MI455X (gfx1250) — compile-verified
//
#include <hip/hip_runtime.h>

typedef float v2f __attribute__((ext_vector_type(2)));
typedef float v8f __attribute__((ext_vector_type(8)));

#define DD 256
#define WAVES_PER_BLOCK 8

// One wave computes a 16-row tile: diag(X_tile @ Y_tile^T) via v_wmma_f32_16x16x4_f32,
// row norms via VALU halves + lane^16 exchange. Writes one partial sum per wave.
__global__ __launch_bounds__(256) void simloss_tile_kernel(
    const float* __restrict__ X, const float* __restrict__ Y,
    float* __restrict__ partial, int numTiles)
{
    const int lane = threadIdx.x & 31;
    const int wave = threadIdx.x >> 5;
    const int w = blockIdx.x * WAVES_PER_BLOCK + wave;   // tile index (uniform per wave)
    if (w >= numTiles) return;                           // whole-wave guard: EXEC stays all-1s

    // A 16x4 f32 fragment layout (ISA 7.12.2): lane L -> row (L&15), K = 2*(L>>4) + {0,1}
    const int row  = (w << 4) + (lane & 15);
    const int coff = (lane >> 4) << 1;                   // 0 or 2
    const float* xp = X + (size_t)row * DD + coff;
    const float* yp = Y + (size_t)row * DD + coff;

    v8f acc = {};              // C = X_tile @ Y_tile^T accumulator (only diagonal used)
    float nx2 = 0.f, ny2 = 0.f;

#pragma unroll 8
    for (int k = 0; k < DD; k += 4) {
        float2 xv = *(const float2*)(xp + k);            // global_load_b64
        float2 yv = *(const float2*)(yp + k);            // global_load_b64
        v2f a; a[0] = xv.x; a[1] = xv.y;
        v2f b; b[0] = yv.x; b[1] = yv.y;
        acc = __builtin_amdgcn_wmma_f32_16x16x4_f32(
            /*neg_a=*/false, a, /*neg_b=*/false, b,
            /*c_mod=*/(short)0, acc, /*reuse_a=*/false, /*reuse_b=*/false);
        nx2 = fmaf(xv.x, xv.x, fmaf(xv.y, xv.y, nx2));
        ny2 = fmaf(yv.x, yv.x, fmaf(yv.y, yv.y, ny2));
    }

    // Lane L holds K%4 in {0,1} (L<16) or {2,3} (L>=16) of row L&15: pair-sum completes norms.
    nx2 += __shfl_xor(nx2, 16, 32);
    ny2 += __shfl_xor(ny2, 16, 32);

    // Diagonal extraction from C layout: lanes 0-7 -> diag i=lane (VGPR=lane),
    // lanes 24-31 -> diag i=lane-16 (VGPR=lane-24). Other lanes contribute 0.
    const int v7 = lane & 7;
    float dsel = acc[0];
    if (v7 == 1) dsel = acc[1];
    if (v7 == 2) dsel = acc[2];
    if (v7 == 3) dsel = acc[3];
    if (v7 == 4) dsel = acc[4];
    if (v7 == 5) dsel = acc[5];
    if (v7 == 6) dsel = acc[6];
    if (v7 == 7) dsel = acc[7];

    float contrib = 0.f;
    if ((lane < 8) || (lane >= 24)) {
        float nx = fmaxf(sqrtf(nx2), 1e-12f);            // F.normalize eps clamp on the norm
        float ny = fmaxf(sqrtf(ny2), 1e-12f);
        contrib = 1.0f - fabsf(dsel) / (nx * ny);
    }

    // wave32 butterfly reduction
    contrib += __shfl_xor(contrib, 1,  32);
    contrib += __shfl_xor(contrib, 2,  32);
    contrib += __shfl_xor(contrib, 4,  32);
    contrib += __shfl_xor(contrib, 8,  32);
    contrib += __shfl_xor(contrib, 16, 32);

    if (lane == 0) partial[w] = contrib;
}

// Deterministic final reduction: fixed-order strided serial sums + LDS tree.
__global__ __launch_bounds__(256) void simloss_reduce_kernel(
    const float* __restrict__ partial, float* __restrict__ out, int n)
{
    __shared__ float s[256];
    float acc = 0.f;
    for (int j = threadIdx.x; j < n; j += 256) acc += partial[j];
    s[threadIdx.x] = acc;
    __syncthreads();
#pragma unroll
    for (int off = 128; off > 0; off >>= 1) {
        if ((int)threadIdx.x < off) s[threadIdx.x] += s[threadIdx.x + off];
        __syncthreads();
    }
    if (threadIdx.x == 0) out[0] = s[0];
}

extern "C" void kernel_launch(void* const* d_in, const int* in_sizes, int n_in,
                              void* d_out, int out_size, void* d_ws, size_t ws_size,
                              hipStream_t stream) {
    const float* X = (const float*)d_in[0];
    const float* Y = (const float*)d_in[1];
    float* out = (float*)d_out;
    float* partial = (float*)d_ws;

    const int total = in_sizes[0];        // N * 256
    const int N = total / DD;             // 131072
    const int numTiles = N >> 4;          // 8192 tiles of 16 rows

    const int blocks = (numTiles + WAVES_PER_BLOCK - 1) / WAVES_PER_BLOCK;
    hipLaunchKernelGGL(simloss_tile_kernel, dim3(blocks), dim3(WAVES_PER_BLOCK * 32), 0, stream,
                       X, Y, partial, numTiles);
    hipLaunchKernelGGL(simloss_reduce_kernel, dim3(1), dim3(256), 0, stream,
                       partial, out, numTiles);
}